// MaskedWeight_56822417326821
// MI455X (gfx1250) — compile-verified
//
#include <hip/hip_runtime.h>
#include <cmath>

// ---------------------------------------------------------------------------
// MaskedWeight on MI455X (gfx1250), wave32, WMMA f32 16x16x4.
//
// out  = inputs @ w.T + bias + cond @ weight_context          (2x f32 GEMM)
// comb = logsumexp reformulated as log( expG @ expGrad ) with rank-1 max
//        stabilization -> 512 small 64x64x64 f32 GEMMs on the WMMA pipes.
// ---------------------------------------------------------------------------

typedef __attribute__((ext_vector_type(2))) float v2f;
typedef __attribute__((ext_vector_type(8))) float v8f;

#define OUT_F 1024
#define IN_F  1024
#define CTX_F 1024
#define BATCH 32
#define DIMB  16

__device__ __forceinline__ v8f wmma_f32(v2f a, v2f b, v8f c) {
  // 8 args: (neg_a, A, neg_b, B, c_mod, C, reuse_a, reuse_b)
  return __builtin_amdgcn_wmma_f32_16x16x4_f32(
      false, a, false, b, (short)0, c, false, false);
}

// ---------------------------------------------------------------------------
// Kernel A: per-row masking, sum-of-squares reduction, normalization.
//   w_norm[r,c]  = exp(diag[r]) * w_raw[r,c] / sqrt(wsn[r])
//   logterm[r]   = diag[r] - 0.5*log(wsn[r])   (used to build g in kernel C)
// One block (256 threads) per row.
// ---------------------------------------------------------------------------
__global__ void rowprep_kernel(const float* __restrict__ weight,
                               const float* __restrict__ diagw,
                               float* __restrict__ w_norm,
                               float* __restrict__ logterm) {
  const int r  = blockIdx.x;
  const int t  = threadIdx.x;
  const int rb = r >> 6;

  __shared__ float wraw[IN_F];
  __shared__ float red[256];
  __shared__ float bc;

  float sq = 0.f;
  for (int c = t; c < IN_F; c += 256) {
    const int cb = c >> 6;
    const float wv  = weight[r * IN_F + c];
    const float val = (cb == rb) ? __expf(wv) : ((cb < rb) ? wv : 0.f);
    wraw[c] = val;
    sq += val * val;
  }
  red[t] = sq;
  __syncthreads();
  for (int s = 128; s > 0; s >>= 1) {
    if (t < s) red[t] += red[t + s];
    __syncthreads();
  }
  if (t == 0) {
    const float wsn = red[0];
    const float dw  = diagw[r];
    bc = __expf(dw) * rsqrtf(wsn);
    logterm[r] = dw - 0.5f * __logf(wsn);
  }
  __syncthreads();
  const float scale = bc;
  for (int c = t; c < IN_F; c += 256)
    w_norm[r * IN_F + c] = scale * wraw[c];
}

// ---------------------------------------------------------------------------
// Kernel B: out = inputs @ w_norm.T + cond @ wctx + bias via WMMA f32.
// One wave32 per 16x16 output tile; grid (64 n-tiles, 2 m-tiles).
// A(16x4):  lane = M + 16*(K>=2), reg = K&1
// B(4x16):  lane = N + 16*(K>=2), reg = K&1
// C(16x16): lane = N + 16*(M>=8), reg = M&7
// ---------------------------------------------------------------------------
__global__ void out_gemm_kernel(const float* __restrict__ inputs,
                                const float* __restrict__ cond,
                                const float* __restrict__ w_norm,
                                const float* __restrict__ wctx,
                                const float* __restrict__ bias,
                                float* __restrict__ out) {
  const int lane = threadIdx.x;          // 0..31 (one wave)
  const int nt   = blockIdx.x;           // 0..63
  const int mt   = blockIdx.y;           // 0..1
  const int mA   = lane & 15;            // A row within tile
  const int kb   = (lane >> 4) << 1;     // K base: 0 or 2
  const int nB   = lane & 15;            // B col within tile

  v8f acc = {};

  // GEMM1: B[K][N] = w_norm[n0+N][K]  (contiguous in K -> coalesced-ish)
  {
    const float* ap = inputs + (mt * 16 + mA) * IN_F + kb;
    const float* bp = w_norm + (nt * 16 + nB) * IN_F + kb;
    for (int k0 = 0; k0 < IN_F; k0 += 4) {
      v2f a; a.x = ap[k0]; a.y = ap[k0 + 1];
      v2f b; b.x = bp[k0]; b.y = bp[k0 + 1];
      acc = wmma_f32(a, b, acc);
    }
  }
  // GEMM2: B[K][N] = wctx[k0+K][n0+N]  (row-major CTX x OUT)
  {
    const float* ap = cond + (mt * 16 + mA) * CTX_F + kb;
    const float* bp = wctx + nt * 16 + nB;
    for (int k0 = 0; k0 < CTX_F; k0 += 4) {
      v2f a; a.x = ap[k0]; a.y = ap[k0 + 1];
      v2f b;
      b.x = bp[(size_t)(k0 + kb) * OUT_F];
      b.y = bp[(size_t)(k0 + kb + 1) * OUT_F];
      acc = wmma_f32(a, b, acc);
    }
  }

  const int mBase = (lane >> 4) << 3;
  const int n     = nt * 16 + (lane & 15);
  const float bv  = bias[n];
  for (int v = 0; v < 8; ++v) {
    const int m = mt * 16 + mBase + v;
    out[m * OUT_F + n] = acc[v] + bv;
  }
}

// ---------------------------------------------------------------------------
// Kernel C: comb[b,d] = m1 + m2^T + log( expG(64x64) @ expGrad(64x64) ).
//   g[d,i,k]        = logterm[d*64+i] + weight[d*64+i][d*64+k]
//   expG[i][k]      = exp(g - m1[i]),  m1[i] = max_k g
//   expGrad[k][j]   = exp(grad[b,d,k,j] - m2[j]),  m2[j] = max_k grad
// One 256-thread block (8 wave32) per (b,d); tiles staged in LDS (stride 66).
// ---------------------------------------------------------------------------
__global__ void comb_kernel(const float* __restrict__ grad,
                            const float* __restrict__ weight,
                            const float* __restrict__ logterm,
                            float* __restrict__ comb) {
  const int d = blockIdx.x;   // 0..15
  const int b = blockIdx.y;   // 0..31
  const int t = threadIdx.x;  // 0..255
  const int STR = 66;         // padded LDS row stride (even, avoids aligned 64)

  __shared__ float eg[64 * 66];   // expG  [i][k]
  __shared__ float er[64 * 66];   // expGrad [k][j]
  __shared__ float m1s[64];
  __shared__ float m2s[64];

  const float* gblk = grad + (size_t)((b * DIMB + d) * 64) * 64;   // [k][j]
  const float* wblk = weight + (size_t)(d * 64) * IN_F + d * 64;   // [i][k]

  if (t < 64) {                              // column max of grad block
    float m = -3.4e38f;
    for (int k = 0; k < 64; ++k) m = fmaxf(m, gblk[k * 64 + t]);
    m2s[t] = m;
  } else if (t < 128) {                      // row max of g block
    const int i = t - 64;
    const float* wr = wblk + (size_t)i * IN_F;
    float m = -3.4e38f;
    for (int k = 0; k < 64; ++k) m = fmaxf(m, wr[k]);
    m1s[i] = logterm[d * 64 + i] + m;
  }
  __syncthreads();

  for (int idx = t; idx < 64 * 64; idx += 256) {
    const int k = idx >> 6, j = idx & 63;
    er[k * STR + j] = __expf(gblk[k * 64 + j] - m2s[j]);
  }
  for (int idx = t; idx < 64 * 64; idx += 256) {
    const int i = idx >> 6, k = idx & 63;
    const float g = logterm[d * 64 + i] + wblk[(size_t)i * IN_F + k];
    eg[i * STR + k] = __expf(g - m1s[i]);
  }
  __syncthreads();

  const int wave  = t >> 5;
  const int lane  = t & 31;
  const int mA    = lane & 15;
  const int kb    = (lane >> 4) << 1;
  const int nB    = lane & 15;
  const int mBase = (lane >> 4) << 3;

  for (int tile = wave; tile < 16; tile += 8) {   // wave-uniform: EXEC stays full
    const int i0 = (tile >> 2) * 16;
    const int j0 = (tile & 3) * 16;
    v8f acc = {};
    const float* ap = &eg[(i0 + mA) * STR + kb];
    const float* bp = &er[kb * STR + j0 + nB];
    for (int kk = 0; kk < 16; ++kk) {
      v2f a; a.x = ap[kk * 4];           a.y = ap[kk * 4 + 1];
      v2f b; b.x = bp[(kk * 4) * STR];   b.y = bp[(kk * 4 + 1) * STR];
      acc = wmma_f32(a, b, acc);
    }
    for (int v = 0; v < 8; ++v) {
      const int i = i0 + mBase + v;
      const int j = j0 + (lane & 15);
      comb[((size_t)(b * DIMB + d) * 64 + i) * 64 + j] =
          m1s[i] + m2s[j] + __logf(acc[v]);
    }
  }
}

// ---------------------------------------------------------------------------
extern "C" void kernel_launch(void* const* d_in, const int* in_sizes, int n_in,
                              void* d_out, int out_size, void* d_ws, size_t ws_size,
                              hipStream_t stream) {
  (void)in_sizes; (void)n_in; (void)out_size; (void)ws_size;
  const float* inputs = (const float*)d_in[0];
  const float* grad   = (const float*)d_in[1];
  const float* cond   = (const float*)d_in[2];
  const float* weight = (const float*)d_in[3];
  const float* diagw  = (const float*)d_in[4];
  const float* bias   = (const float*)d_in[5];
  const float* wctx   = (const float*)d_in[6];

  float* out  = (float*)d_out;                 // (32, 1024)
  float* comb = out + BATCH * OUT_F;           // (32, 16, 64, 64)

  float* w_norm  = (float*)d_ws;               // 1024*1024 floats
  float* logterm = w_norm + (size_t)OUT_F * IN_F;  // 1024 floats

  rowprep_kernel<<<dim3(OUT_F), dim3(256), 0, stream>>>(weight, diagw, w_norm, logterm);
  out_gemm_kernel<<<dim3(64, 2), dim3(32), 0, stream>>>(inputs, cond, w_norm, wctx, bias, out);
  comb_kernel<<<dim3(DIMB, BATCH), dim3(256), 0, stream>>>(grad, weight, logterm, comb);
}